// AutoEncIndex_33887291965861
// MI455X (gfx1250) — compile-verified
//
#include <hip/hip_runtime.h>
#include <stdint.h>

#define HW 32768
#define J  1024

typedef int v4i __attribute__((ext_vector_type(4)));
#define AS1 __attribute__((address_space(1)))
#define AS3 __attribute__((address_space(3)))

// ---- CDNA5 async global->LDS path (ASYNCcnt-tracked), with safe fallback ----
#if defined(__HIP_DEVICE_COMPILE__) && __has_builtin(__builtin_amdgcn_global_load_async_to_lds_b128)
#define USE_ASYNC_LDS 1
#else
#define USE_ASYNC_LDS 0
#endif

#if defined(__HIP_DEVICE_COMPILE__) && __has_builtin(__builtin_amdgcn_s_wait_asynccnt)
#define WAIT_ASYNC() __builtin_amdgcn_s_wait_asynccnt(0)
#elif defined(__HIP_DEVICE_COMPILE__)
#define WAIT_ASYNC() asm volatile("s_wait_asynccnt 0" ::: "memory")
#else
#define WAIT_ASYNC() do {} while (0)
#endif

// -------------------------------------------------------------------------
// Kernel 1: per-row argmax of logits (mat [+ noise]) with lowest-index
// tie-break, and zero-fill of the output row. One 256-thread block per row.
// -------------------------------------------------------------------------
__global__ void __launch_bounds__(256) k_rowmax_zero(
    const float* __restrict__ mat, const float* __restrict__ noise,
    const int* __restrict__ f_noise, const int* __restrict__ f_train,
    float* __restrict__ out, int* __restrict__ sel)
{
    __shared__ float sm[J];
    __shared__ float sn[J];
    __shared__ float rv[8];
    __shared__ int   ri[8];

    const int r = blockIdx.x;
    const int t = threadIdx.x;                 // 0..255
    const size_t rowoff = (size_t)r * J;
    const bool addn = (f_noise[0] != 0) && (f_train[0] != 0);

#if USE_ASYNC_LDS
    // 256 lanes x 16B = one 4KB row per instruction, DMA'd into LDS.
    __builtin_amdgcn_global_load_async_to_lds_b128(
        (AS1 v4i*)(mat + rowoff + 4 * t),
        (AS3 v4i*)(&sm[4 * t]), 0, 0);
    __builtin_amdgcn_global_load_async_to_lds_b128(
        (AS1 v4i*)(noise + rowoff + 4 * t),
        (AS3 v4i*)(&sn[4 * t]), 0, 0);
    WAIT_ASYNC();                 // each wave drains its own ASYNCcnt
#else
    {
        const float4 m4 = ((const float4*)(mat + rowoff))[t];
        const float4 n4 = ((const float4*)(noise + rowoff))[t];
        ((float4*)sm)[t] = m4;
        ((float4*)sn)[t] = n4;
    }
#endif
    __syncthreads();              // then cross-wave visibility via barrier

    // zero the output row (straight-through: (0 - m) + m == 0 exactly)
    ((float4*)(out + rowoff))[t] = make_float4(0.f, 0.f, 0.f, 0.f);

    float bv = -3.0e38f;
    int   bi = 0;
    for (int i = t; i < J; i += 256) {
        const float v = addn ? (sm[i] + sn[i]) : sm[i];
        if (v > bv) { bv = v; bi = i; }
    }
    // wave32 shuffle reduction (gfx1250 is wave32-only)
    for (int off = 16; off > 0; off >>= 1) {
        const float ov = __shfl_down(bv, off, 32);
        const int   oi = __shfl_down(bi, off, 32);
        if (ov > bv || (ov == bv && oi < bi)) { bv = ov; bi = oi; }
    }
    const int wave = t >> 5, lane = t & 31;
    if (lane == 0) { rv[wave] = bv; ri[wave] = bi; }
    __syncthreads();
    if (t == 0) {
        float v = rv[0];
        int idx = ri[0];
        for (int w = 1; w < 8; ++w)
            if (rv[w] > v || (rv[w] == v && ri[w] < idx)) { v = rv[w]; idx = ri[w]; }
        sel[r] = idx;
    }
}

// -------------------------------------------------------------------------
// Kernel 2: full descending argsort (lowest index first on ties) of the
// first J rows via in-LDS bitonic sort on 64-bit (value,index) keys.
// -------------------------------------------------------------------------
__global__ void __launch_bounds__(1024) k_sort_rows(
    const float* __restrict__ mat, const float* __restrict__ noise,
    const int* __restrict__ f_noise, const int* __restrict__ f_train,
    int* __restrict__ sorted_idx)
{
    __shared__ unsigned long long key[J];
    const int r = blockIdx.x;
    const int t = threadIdx.x;                 // 0..1023
    const size_t rowoff = (size_t)r * J;
    const bool addn = (f_noise[0] != 0) && (f_train[0] != 0);

    const float v = addn ? (mat[rowoff + t] + noise[rowoff + t]) : mat[rowoff + t];
    unsigned u = __float_as_uint(v);
    u = (u & 0x80000000u) ? ~u : (u | 0x80000000u);  // order-preserving map
    // ascending sort of key <=> descending value, ties -> ascending index
    key[t] = ((unsigned long long)(~u) << 32) | (unsigned)t;
    __syncthreads();

    for (int k = 2; k <= J; k <<= 1) {
        for (int j = k >> 1; j > 0; j >>= 1) {
            const int ixj = t ^ j;
            if (ixj > t) {
                const bool up = ((t & k) == 0);
                const unsigned long long a = key[t];
                const unsigned long long b = key[ixj];
                if (up ? (a > b) : (a < b)) { key[t] = b; key[ixj] = a; }
            }
            __syncthreads();
        }
    }
    sorted_idx[rowoff + t] = (int)(key[t] & 0xFFFFFFFFull);
}

// -------------------------------------------------------------------------
// Kernel 3: sequential greedy joint assignment over the first J rows.
// Early exit makes the expected probe count ~ J * H(J) (tiny).
// -------------------------------------------------------------------------
__global__ void __launch_bounds__(1024) k_greedy(
    const int* __restrict__ sorted_idx, int* __restrict__ sel)
{
    __shared__ int ind[J];
    const int t = threadIdx.x;
    ind[t] = 1;
    __syncthreads();
    if (t == 0) {
        for (int r = 0; r < J; ++r) {
            const int* cand = sorted_idx + (size_t)r * J;
            int s = cand[0];
            for (int c = 0; c <= r; ++c) {
                const int jj = cand[c];
                if (ind[jj]) { s = jj; break; }   // first available always exists
            }
            ind[s] = 0;
            sel[r] = s;
        }
    }
}

// -------------------------------------------------------------------------
// Kernel 4: write the straight-through one-hot entry per row.
// -------------------------------------------------------------------------
__global__ void __launch_bounds__(256) k_fixup(
    const float* __restrict__ mat, const int* __restrict__ sel,
    const int* __restrict__ f_train, float* __restrict__ out)
{
    const int r = blockIdx.x * blockDim.x + threadIdx.x;
    if (r >= HW) return;
    const int s = sel[r];
    const size_t off = (size_t)r * J + (size_t)s;
    const float m = mat[off];
    out[off] = (f_train[0] != 0) ? ((1.0f - m) + m) : 1.0f;
}

extern "C" void kernel_launch(void* const* d_in, const int* in_sizes, int n_in,
                              void* d_out, int out_size, void* d_ws, size_t ws_size,
                              hipStream_t stream) {
    (void)in_sizes; (void)n_in; (void)out_size; (void)ws_size;
    const float* mat     = (const float*)d_in[0];
    const float* noise   = (const float*)d_in[1];
    const int*   f_noise = (const int*)d_in[2];
    const int*   f_train = (const int*)d_in[3];
    float* out = (float*)d_out;

    int* sel        = (int*)d_ws;                                     // HW ints
    int* sorted_idx = (int*)((char*)d_ws + (size_t)HW * sizeof(int)); // J*J ints

    k_rowmax_zero<<<HW, 256, 0, stream>>>(mat, noise, f_noise, f_train, out, sel);
    k_sort_rows<<<J, J, 0, stream>>>(mat, noise, f_noise, f_train, sorted_idx);
    k_greedy<<<1, J, 0, stream>>>(sorted_idx, sel);
    k_fixup<<<(HW + 255) / 256, 256, 0, stream>>>(mat, sel, f_train, out);
}